// baseline_attention_46394236731717
// MI455X (gfx1250) — compile-verified
//
#include <hip/hip_runtime.h>
#include <math.h>

typedef __attribute__((ext_vector_type(2))) float v2f;
typedef __attribute__((ext_vector_type(8))) float v8f;

#define B_ 16
#define D_ 2048
#define V_ 504
#define R_ 190
#define H_ 32
#define MR_ 6
#define G_ 2000
#define NH_ 4
#define HD_ 8

// flat d_out offsets (reference tuple return order)
#define OFF_VPOT   0
#define OFF_NORM   (B_*V_)                    /* 8064  */
#define OFF_VMAX   (OFF_NORM + B_)            /* 8080  */
#define OFF_VRN    (OFF_VMAX + B_*V_)         /* 16144 */
#define OFF_VRMAX  (OFF_VRN + B_*V_*MR_)      /* 64528 */
#define OFF_VRMAXI (OFF_VRMAX + B_*V_*MR_)    /* 112912*/

__device__ __forceinline__ v8f wmma_f32_16x16x4(v2f a, v2f b, v8f c) {
#if defined(__gfx1250__)
  // (neg_a, A, neg_b, B, c_mod, C, reuse_a, reuse_b)
  return __builtin_amdgcn_wmma_f32_16x16x4_f32(false, a, false, b, (short)0, c, false, false);
#else
  (void)a; (void)b;
  return c;
#endif
}

// C(16xN) = A(16xK) @ W(KxN) + bias ; one wave per 16-col tile.
// N,K compile-time so all W row strides fold into constant instruction offsets.
// 32-bit A 16x4 layout: VGPR0 lanes0-15 K=0 / lanes16-31 K=2 ; VGPR1 K=1/K=3.
// B 4x16 mirrors: VGPR0 rows K=0 (lanes0-15, N=lane) / K=2 (lanes16-31); VGPR1 K=1/K=3.
template <int N, int K>
__global__ void gemm16_wmma_kernel(const float* __restrict__ A,
                                   const float* __restrict__ W,
                                   const float* __restrict__ bias,
                                   float* __restrict__ out) {
  const int lane = threadIdx.x & 31;
  const int m    = lane & 15;
  const int half = lane >> 4;
  const int n0   = blockIdx.x * 16;
  const int ncol = n0 + m;
  const int ncl  = ncol < N ? ncol : (N - 1);   // clamp OOB loads (tail tile)
  v8f c = {0.f,0.f,0.f,0.f,0.f,0.f,0.f,0.f};
  const float* arow = A + (size_t)m * K + 2*half;     // consecutive pair -> b64 load
  const float* wp   = W + (size_t)(2*half) * N + ncl; // row-strided, const offsets
  for (int kk = 0; kk < K; kk += 32) {
    // one prefetch instruction: 32 lanes fan across W rows kk+64..kk+95 (two k-steps ahead)
    {
      int pr = kk + 64 + lane;
      if (pr > K - 1) pr = K - 1;
      __builtin_prefetch(W + (size_t)pr * N + n0, 0, 3);
    }
    #pragma unroll
    for (int dk = 0; dk < 32; dk += 4) {
      v2f a, b;
      a.x = arow[kk + dk];
      a.y = arow[kk + dk + 1];
      b.x = wp[(size_t)(kk + dk)     * N];
      b.y = wp[(size_t)(kk + dk + 1) * N];
      c = wmma_f32_16x16x4(a, b, c);
    }
  }
  if (ncol < N) {
    float bv = bias[ncol];
    #pragma unroll
    for (int i = 0; i < 8; ++i)        // lane holds rows i+8*half, col ncol
      out[(size_t)(i + 8*half) * N + ncol] = c[i] + bv;
  }
}

// 3 MHA layers over x(v) = node[b, ridx[s], :] gathered tile; one block per v.
__global__ void mha_kernel(const float* __restrict__ node,     // B x (R*H)
                           const float* __restrict__ in_w,     // L x 96 x 32
                           const float* __restrict__ in_b,     // L x 96
                           const float* __restrict__ out_w,    // L x 32 x 32
                           const float* __restrict__ out_b,    // L x 32
                           const int* __restrict__ verb_roles, // V x MR
                           float* __restrict__ gx)             // V*MR*B*H
{
  __shared__ float xs[MR_][B_][H_];
  __shared__ float qkv[MR_][B_][3*H_];
  __shared__ float att[MR_][B_][H_];
  __shared__ int ridx[MR_];
  __shared__ int pmask[MR_];
  const int v = blockIdx.x;
  const int tid = threadIdx.x;
  const int NT = blockDim.x;
  if (tid < MR_) {
    int r = verb_roles[v*MR_ + tid];
    pmask[tid] = (r >= 0);
    ridx[tid]  = (r < 0) ? 0 : r;
  }
  __syncthreads();
  for (int i = tid; i < MR_*B_*H_; i += NT) {
    int s = i / (B_*H_);
    int rem = i - s*(B_*H_);
    int b = rem / H_;
    int h = rem & (H_-1);
    xs[s][b][h] = node[(size_t)b*(R_*H_) + ridx[s]*H_ + h];
  }
  __syncthreads();
  const float scale = 0.35355339059327373f;   // 1/sqrt(hd=8)
  for (int l = 0; l < 3; ++l) {
    const float* iw = in_w + (size_t)l*96*32;
    const float* ib = in_b + l*96;
    for (int t = tid; t < MR_*B_*96; t += NT) {
      int s = t / (B_*96);
      int rem = t - s*(B_*96);
      int b = rem / 96;
      int f = rem - b*96;
      float acc = ib[f];
      const float* wrow = iw + f*32;
      #pragma unroll
      for (int h = 0; h < 32; ++h) acc += xs[s][b][h] * wrow[h];
      qkv[s][b][f] = acc;
    }
    __syncthreads();
    for (int t = tid; t < MR_*B_*NH_; t += NT) {
      int s    = t % MR_;
      int bh   = t / MR_;
      int head = bh & 3;
      int b    = bh >> 2;
      const int qo = head*HD_;
      float sc[MR_];
      float mx = -INFINITY;
      for (int tt = 0; tt < MR_; ++tt) {
        float d = 0.f;
        #pragma unroll
        for (int k = 0; k < HD_; ++k)
          d += qkv[s][b][qo+k] * qkv[tt][b][32 + qo + k];
        d *= scale;
        d = pmask[tt] ? d : -INFINITY;
        sc[tt] = d;
        mx = fmaxf(mx, d);
      }
      float ssum = 0.f;
      for (int tt = 0; tt < MR_; ++tt) { float e = expf(sc[tt]-mx); sc[tt]=e; ssum += e; }
      float inv = 1.f / ssum;
      #pragma unroll
      for (int k = 0; k < HD_; ++k) {
        float o = 0.f;
        for (int tt = 0; tt < MR_; ++tt) o += sc[tt] * qkv[tt][b][64 + qo + k];
        att[s][b][qo + k] = o * inv;
      }
    }
    __syncthreads();
    const float* ow = out_w + (size_t)l*32*32;
    const float* ob = out_b + l*32;
    for (int t = tid; t < MR_*B_*H_; t += NT) {
      int s = t / (B_*H_);
      int rem = t - s*(B_*H_);
      int b = rem / H_;
      int e = rem & 31;
      float acc = ob[e];
      const float* wrow = ow + e*32;
      #pragma unroll
      for (int h = 0; h < 32; ++h) acc += att[s][b][h] * wrow[h];
      xs[s][b][e] = acc;
    }
    __syncthreads();
  }
  const float* xflat = &xs[0][0][0];
  for (int i = tid; i < MR_*B_*H_; i += NT)
    gx[(size_t)v*(MR_*B_*H_) + i] = xflat[i];
}

// One wave per (v,s): logits(16x2000) = X(16x32) @ W_n[r](32x2000) + b_n[r],
// fused masked online LSE / max / argmax over G per row b.
__global__ void logits_lse_kernel(const float* __restrict__ gx,
                                  const float* __restrict__ W_n,  // R x H x G
                                  const float* __restrict__ b_n,  // R x G
                                  const int* __restrict__ verb_roles,
                                  const int* __restrict__ group_len,
                                  float* __restrict__ out)
{
  const int vs = blockIdx.x;
  const int v  = vs / MR_;
  const int s  = vs - v*MR_;
  const int lane = threadIdx.x & 31;
  const int m    = lane & 15;
  const int half = lane >> 4;
  const int r = verb_roles[v*MR_ + s];

  if (r < 0) {                  // pos_mask false -> outputs forced to 0
    if (m == 0) {
      #pragma unroll
      for (int i = 0; i < 8; ++i) {
        int b = i + 8*half;
        size_t idx = (size_t)b*(V_*MR_) + (size_t)v*MR_ + s;
        out[OFF_VRN + idx]    = 0.f;
        out[OFF_VRMAX + idx]  = 0.f;
        out[OFF_VRMAXI + idx] = 0.f;
      }
    }
    return;
  }
  const int glen = group_len[r];
  const float* Wn = W_n + (size_t)r * H_ * G_;
  const float* bn = b_n + (size_t)r * G_;
  const float* X  = gx + ((size_t)v*MR_ + s) * (B_*H_);

  v2f a[8];
  #pragma unroll
  for (int j = 0; j < 8; ++j) {
    a[j].x = X[m*H_ + 4*j + 2*half];
    a[j].y = X[m*H_ + 4*j + 2*half + 1];
  }
  float mrun[8]; float srun[8]; int arun[8];
  #pragma unroll
  for (int i = 0; i < 8; ++i) { mrun[i] = -INFINITY; srun[i] = 0.f; arun[i] = 0; }

  for (int t = 0; t < G_/16; ++t) {
    const int g = t*16 + m;                  // this lane's column
    // prefetch next tile: 32 lanes fan across the 32 K-rows of W_n[r]
    if (t + 1 < G_/16)
      __builtin_prefetch(Wn + (size_t)lane * G_ + (t + 1) * 16, 0, 3);
    v8f c = {0.f,0.f,0.f,0.f,0.f,0.f,0.f,0.f};
    const float* Bp = Wn + g;
    #pragma unroll
    for (int j = 0; j < 8; ++j) {
      v2f bb;
      bb.x = Bp[(size_t)(4*j + 2*half)     * G_];
      bb.y = Bp[(size_t)(4*j + 2*half + 1) * G_];
      c = wmma_f32_16x16x4(a[j], bb, c);
    }
    if (g < glen) {
      const float bias = bn[g];
      #pragma unroll
      for (int i = 0; i < 8; ++i) {
        float val = c[i] + bias;
        if (val > mrun[i]) {                 // strict > keeps first-index argmax within lane
          srun[i] = srun[i] * expf(mrun[i] - val) + 1.f;
          mrun[i] = val;
          arun[i] = g;
        } else {
          srun[i] += expf(val - mrun[i]);
        }
      }
    }
  }
  // butterfly across the 16 lanes of each half (rows 0-7 | rows 8-15)
  #pragma unroll
  for (int off = 8; off >= 1; off >>= 1) {
    #pragma unroll
    for (int i = 0; i < 8; ++i) {
      float m2 = __shfl_xor(mrun[i], off, 32);
      float s2 = __shfl_xor(srun[i], off, 32);
      int   a2 = __shfl_xor(arun[i], off, 32);
      float mn = fmaxf(mrun[i], m2);
      srun[i] = srun[i]*expf(mrun[i]-mn) + s2*expf(m2-mn);
      arun[i] = (m2 > mrun[i] || (m2 == mrun[i] && a2 < arun[i])) ? a2 : arun[i];
      mrun[i] = mn;
    }
  }
  if (m == 0) {
    #pragma unroll
    for (int i = 0; i < 8; ++i) {
      int b = i + 8*half;
      size_t idx = (size_t)b*(V_*MR_) + (size_t)v*MR_ + s;
      out[OFF_VRN + idx]    = mrun[i] + logf(srun[i]);
      out[OFF_VRMAX + idx]  = mrun[i];
      out[OFF_VRMAXI + idx] = (float)arun[i];
    }
  }
}

// per-b: v_marginal = vp + sum_s vrn ; norm = LSE_v ; v_max = vp + sum_s vr_max
__global__ void finalize_kernel(float* __restrict__ out) {
  const int b = blockIdx.x;
  const int tid = threadIdx.x;
  __shared__ float vm[V_];
  __shared__ float red[256];
  for (int v = tid; v < V_; v += blockDim.x) {
    float vp = out[OFF_VPOT + b*V_ + v];
    float slse = 0.f, smax = 0.f;
    size_t base = (size_t)b*(V_*MR_) + (size_t)v*MR_;
    #pragma unroll
    for (int s = 0; s < MR_; ++s) {
      slse += out[OFF_VRN + base + s];
      smax += out[OFF_VRMAX + base + s];
    }
    vm[v] = vp + slse;
    out[OFF_VMAX + b*V_ + v] = vp + smax;
  }
  __syncthreads();
  float pmax = -INFINITY;
  for (int v = tid; v < V_; v += blockDim.x) pmax = fmaxf(pmax, vm[v]);
  red[tid] = pmax;
  __syncthreads();
  for (int off = 128; off >= 1; off >>= 1) {
    if (tid < off) red[tid] = fmaxf(red[tid], red[tid + off]);
    __syncthreads();
  }
  float gmax = red[0];
  __syncthreads();
  float psum = 0.f;
  for (int v = tid; v < V_; v += blockDim.x) psum += expf(vm[v] - gmax);
  red[tid] = psum;
  __syncthreads();
  for (int off = 128; off >= 1; off >>= 1) {
    if (tid < off) red[tid] += red[tid + off];
    __syncthreads();
  }
  if (tid == 0) out[OFF_NORM + b] = gmax + logf(red[0]);
}

extern "C" void kernel_launch(void* const* d_in, const int* in_sizes, int n_in,
                              void* d_out, int out_size, void* d_ws, size_t ws_size,
                              hipStream_t stream) {
  (void)in_sizes; (void)n_in; (void)out_size; (void)ws_size;
  const float* rep        = (const float*)d_in[0];
  const float* W_v        = (const float*)d_in[1];
  const float* b_v        = (const float*)d_in[2];
  const float* W_r        = (const float*)d_in[3];
  const float* b_r        = (const float*)d_in[4];
  const float* attn_in_w  = (const float*)d_in[5];
  const float* attn_in_b  = (const float*)d_in[6];
  const float* attn_out_w = (const float*)d_in[7];
  const float* attn_out_b = (const float*)d_in[8];
  const float* W_n        = (const float*)d_in[9];
  const float* b_n        = (const float*)d_in[10];
  const int*   verb_roles = (const int*)d_in[11];
  const int*   group_len  = (const int*)d_in[12];
  float* out = (float*)d_out;

  float* node = (float*)d_ws;                 // B x (R*H) = 16 x 6080
  float* gx   = node + (size_t)B_ * R_ * H_;  // V x MR x B x H

  // v_potential -> d_out[0 .. 8064)
  gemm16_wmma_kernel<V_, D_><<<(V_ + 15)/16, 32, 0, stream>>>(rep, W_v, b_v, out + OFF_VPOT);
  // node -> workspace
  gemm16_wmma_kernel<R_*H_, D_><<<(R_*H_)/16, 32, 0, stream>>>(rep, W_r, b_r, node);
  // 3 MHA layers -> gx
  mha_kernel<<<V_, 128, 0, stream>>>(node, attn_in_w, attn_in_b, attn_out_w, attn_out_b,
                                     verb_roles, gx);
  // logits + masked LSE/max/argmax -> vrn / vr_max / vr_maxi
  logits_lse_kernel<<<V_*MR_, 32, 0, stream>>>(gx, W_n, b_n, verb_roles, group_len, out);
  // norm / v_max
  finalize_kernel<<<B_, 256, 0, stream>>>(out);
}